// SamplingLayer_5360119186054
// MI455X (gfx1250) — compile-verified
//
#include <hip/hip_runtime.h>

// Problem constants (fixed by the reference)
#define LT        64                 // L_tilde (N)
#define L         1024               // dense L (K)
#define M_TOTAL   131072             // 16384 * 8 GEMM rows

// Tiling
#define MT        128                // rows per workgroup
#define KT        64                 // K chunk staged in LDS (double-buffered)
#define NCHUNK    (L / KT)           // 16 (even -> clean pair-unrolled pipeline)
#define A_STRIDE  68                 // padded LDS A row stride (floats): conflict-free ds_load_b64

typedef __attribute__((ext_vector_type(2))) float v2f;
typedef __attribute__((ext_vector_type(4))) float v4f;
typedef __attribute__((ext_vector_type(8))) float v8f;

// Copy 128 contiguous bytes global -> LDS via async engine (8 x b128 per lane),
// using the instruction's immediate offset (applied to both addresses per ISA).
__device__ __forceinline__ void async_cp128(unsigned ldst, unsigned long long ga) {
#define CP(off) asm volatile("global_load_async_to_lds_b128 %0, %1, off offset:" off \
                             :: "v"(ldst), "v"(ga) : "memory")
    CP("0"); CP("16"); CP("32"); CP("48"); CP("64"); CP("80"); CP("96"); CP("112");
#undef CP
}

__global__ __launch_bounds__(256) void sampling_gemm_wmma(
    const float* __restrict__ x,        // [M_TOTAL, L] row-major
    const float* __restrict__ weight,   // [LT]
    float* __restrict__ out)            // [M_TOTAL, LT]
{
    // W tile, K-quad interleaved: lds_w[b][ (q*LT + n)*4 + r ] = W[kc + 4q + r][n].
    // Declared first so all offsets (< 32 KB) fold into DS immediate offsets.
    __shared__ float lds_w[2][KT * LT];         // 2 x 16,384 B
    __shared__ float lds_a[2][MT * A_STRIDE];   // 2 x 34,816 B

    const int tid    = threadIdx.x;
    const int lane   = tid & 31;
    const int wave   = tid >> 5;                // 0..7
    const int laneLo = lane & 15;
    const int laneHi = lane >> 4;               // 0 or 1

    const int mBlock = blockIdx.x * MT;

    // per-thread W generator: one column, 4 K-quads per chunk (16 expf)
    const int   wcol    = tid & 63;
    const float wn      = weight[wcol];
    const int   wq0     = tid >> 6;             // 0..3
    const float inv_var = 1.0f / (10.0f * 10.0f);

    // async A-copy assignment: 2 threads per row, 128 B (32 floats) each
    const int arow  = tid >> 1;                 // 0..127
    const int ahalf = (tid & 1) * 32;           // float offset 0 or 32

    // Prefetch chunk `kc` into buffer `b`: async A tile + regenerated W tile.
    auto issue_chunk = [&](int kc, int b) {
        const float* gsrc = x + (size_t)(mBlock + arow) * L + (kc + ahalf);
        unsigned     ldst = (unsigned)(size_t)&lds_a[b][arow * A_STRIDE + ahalf];
        async_cp128(ldst, (unsigned long long)(size_t)gsrc);
        #pragma unroll
        for (int i = 0; i < 4; ++i) {
            int   q  = wq0 + i * 4;                        // 0..15
            float t0 = (float)(kc + 4 * q + 1);
            v4f   e;
            #pragma unroll
            for (int r = 0; r < 4; ++r) {
                float d = (t0 + (float)r) - wn;
                e[r] = __expf(-d * d * inv_var);
            }
            *(v4f*)&lds_w[b][(q * LT + wcol) * 4] = e;     // ds_store_b128
        }
    };

    v8f acc0 = {}, acc1 = {}, acc2 = {}, acc3 = {};

    // Consume chunk in buffer `b` (b is always a literal -> immediate DS offsets).
    auto mma_chunk = [&](int b) {
        const float* aw = &lds_a[b][(wave * 16 + laneLo) * A_STRIDE + 2 * laneHi];
        const float* bw = &lds_w[b][laneLo * 4 + 2 * laneHi];
        #pragma unroll 4
        for (int k = 0; k < KT; k += 4) {
            v2f a  = *(const v2f*)(aw + k);                // A frag: 16 rows x K pair
            v2f b0 = *(const v2f*)(bw + (k + 0) * LT);     // single b64 per B frag,
            v2f b1 = *(const v2f*)(bw + (k + 1) * LT);     // conflict-free banks
            v2f b2 = *(const v2f*)(bw + (k + 2) * LT);
            v2f b3 = *(const v2f*)(bw + (k + 3) * LT);
            acc0 = __builtin_amdgcn_wmma_f32_16x16x4_f32(false, a, false, b0, (short)0, acc0, false, false);
            acc1 = __builtin_amdgcn_wmma_f32_16x16x4_f32(false, a, false, b1, (short)0, acc1, false, false);
            acc2 = __builtin_amdgcn_wmma_f32_16x16x4_f32(false, a, false, b2, (short)0, acc2, false, false);
            acc3 = __builtin_amdgcn_wmma_f32_16x16x4_f32(false, a, false, b3, (short)0, acc3, false, false);
        }
    };

    // ---- prologue: stage chunk 0 into buffer 0 ----
    issue_chunk(0, 0);
    asm volatile("s_wait_asynccnt 0" ::: "memory");
    __syncthreads();

    // ---- pair-unrolled double-buffered pipeline (buffer index is literal) ----
    for (int c = 0; c < NCHUNK; c += 2) {
        // buffer 0 holds chunk c; prefetch chunk c+1 into buffer 1
        issue_chunk((c + 1) * KT, 1);
        mma_chunk(0);
        asm volatile("s_wait_asynccnt 0" ::: "memory");
        __syncthreads();

        // buffer 1 holds chunk c+1; prefetch chunk c+2 into buffer 0
        if (c + 2 < NCHUNK) issue_chunk((c + 2) * KT, 0);
        mma_chunk(1);
        asm volatile("s_wait_asynccnt 0" ::: "memory");
        __syncthreads();
    }

    // ---- store C: VGPR r holds M=r (lanes 0-15) / M=r+8 (lanes 16-31) ----
    const int m0 = mBlock + wave * 16 + 8 * laneHi;
    float* o = out + (size_t)m0 * LT + laneLo;
    #pragma unroll
    for (int r = 0; r < 8; ++r) {
        float* orow = o + (size_t)r * LT;
        orow[0]  = acc0[r];
        orow[16] = acc1[r];
        orow[32] = acc2[r];
        orow[48] = acc3[r];
    }
}

extern "C" void kernel_launch(void* const* d_in, const int* in_sizes, int n_in,
                              void* d_out, int out_size, void* d_ws, size_t ws_size,
                              hipStream_t stream) {
    (void)in_sizes; (void)n_in; (void)out_size; (void)d_ws; (void)ws_size;
    const float* x      = (const float*)d_in[0];
    const float* weight = (const float*)d_in[1];
    float*       out    = (float*)d_out;

    dim3 grid(M_TOTAL / MT);   // 1024 workgroups
    dim3 block(256);           // 8 wave32
    sampling_gemm_wmma<<<grid, block, 0, stream>>>(x, weight, out);
}